// ImageManifoldHead_51410758533313
// MI455X (gfx1250) — compile-verified
//
#include <hip/hip_runtime.h>

typedef __attribute__((ext_vector_type(2))) float v2f;
typedef __attribute__((ext_vector_type(8))) float v8f;

#define IMG     448
#define PSZ     16
#define NH      28
#define NPATCH  784
#define PDIM    768
#define DMODEL  512
#define DCOORD  8
#define DSPIN   4
#define NBATCH  32
#define NROWS   (NBATCH * NPATCH)   /* 25088 */
#define LN_EPS  1e-5f

// ---------------------------------------------------------------------------
// Kernel 1: x = patchify(images) @ W_patch + b_patch + pos_embed
// One wave per 32x32 output tile (2x2 WMMA register blocking). A fragments
// are gathered straight from images as contiguous float2 (patch-row pixels).
// ---------------------------------------------------------------------------
__global__ __launch_bounds__(128) void patch_embed_wmma(
    const float* __restrict__ images, const float* __restrict__ Wp,
    const float* __restrict__ bp, const float* __restrict__ pos,
    float* __restrict__ x)
{
    const int lane  = threadIdx.x & 31;
    const int wave  = threadIdx.x >> 5;
    const int tileN = blockIdx.x;                 // 0..15  (32 cols each)
    const int tileM = blockIdx.y * 4 + wave;      // 0..783 (32 rows each)
    const int l16   = lane & 15;
    const int khalf = lane >> 4;                  // 0 or 1

    const int mrow0 = tileM * 32 + l16;           // A rows this lane feeds
    const int mrow1 = mrow0 + 16;
    const int ncol0 = tileN * 32 + l16;           // B cols this lane feeds
    const int ncol1 = ncol0 + 16;

    // Patch-gather bases: row = b*784 + (bh*28 + bw)
    int lbase0, lbase1;
    {
        int b  = mrow0 / NPATCH;
        int p  = mrow0 - b * NPATCH;
        int bh = p / NH, bw = p - bh * NH;
        lbase0 = b * (3 * IMG * IMG) + (bh * PSZ) * IMG + (bw * PSZ);
        b  = mrow1 / NPATCH;
        p  = mrow1 - b * NPATCH;
        bh = p / NH; bw = p - bh * NH;
        lbase1 = b * (3 * IMG * IMG) + (bh * PSZ) * IMG + (bw * PSZ);
    }

    v8f acc00 = {}, acc01 = {}, acc10 = {}, acc11 = {};
    #pragma unroll 2
    for (int k0 = 0; k0 < PDIM; k0 += 4) {
        const int ka  = k0 + khalf * 2;           // per-lane K pair (even)
        const int c   = ka >> 8;                  // channel
        const int i   = (ka >> 4) & 15;           // row in patch
        const int j   = ka & 15;                  // col in patch (even)
        const int off = c * (IMG * IMG) + i * IMG + j;

        const v2f a0 = *(const v2f*)(images + lbase0 + off);
        const v2f a1 = *(const v2f*)(images + lbase1 + off);
        v2f b0, b1;
        b0.x = Wp[ka * DMODEL + ncol0];
        b0.y = Wp[(ka + 1) * DMODEL + ncol0];
        b1.x = Wp[ka * DMODEL + ncol1];
        b1.y = Wp[(ka + 1) * DMODEL + ncol1];

        acc00 = __builtin_amdgcn_wmma_f32_16x16x4_f32(false, a0, false, b0, (short)0, acc00, false, false);
        acc01 = __builtin_amdgcn_wmma_f32_16x16x4_f32(false, a0, false, b1, (short)0, acc01, false, false);
        acc10 = __builtin_amdgcn_wmma_f32_16x16x4_f32(false, a1, false, b0, (short)0, acc10, false, false);
        acc11 = __builtin_amdgcn_wmma_f32_16x16x4_f32(false, a1, false, b1, (short)0, acc11, false, false);
    }

    const float bias0 = bp[ncol0];
    const float bias1 = bp[ncol1];
    // 784 = 49*16, so a 16-row subtile never crosses a batch: row%784 is cheap
    const int prowb0 = ((tileM * 2)     % 49) * 16;
    const int prowb1 = ((tileM * 2 + 1) % 49) * 16;

    #pragma unroll
    for (int r = 0; r < 8; ++r) {
        const int m    = r + 8 * khalf;
        const int row0 = tileM * 32 + m;
        const int row1 = row0 + 16;
        const int pr0  = prowb0 + m;
        const int pr1  = prowb1 + m;
        x[(size_t)row0 * DMODEL + ncol0] = acc00[r] + bias0 + pos[(size_t)pr0 * DMODEL + ncol0];
        x[(size_t)row0 * DMODEL + ncol1] = acc01[r] + bias1 + pos[(size_t)pr0 * DMODEL + ncol1];
        x[(size_t)row1 * DMODEL + ncol0] = acc10[r] + bias0 + pos[(size_t)pr1 * DMODEL + ncol0];
        x[(size_t)row1 * DMODEL + ncol1] = acc11[r] + bias1 + pos[(size_t)pr1 * DMODEL + ncol1];
    }
}

// ---------------------------------------------------------------------------
// Kernel 2: coords/spinor projections, K0 modulation, residual, LayerNorm.
// One wave32 per row; cross-lane reductions via shfl_xor. In-place on x.
// ---------------------------------------------------------------------------
__global__ __launch_bounds__(256) void manifold_ln(
    const float* __restrict__ x,
    const float* __restrict__ Wc, const float* __restrict__ bc,
    const float* __restrict__ Ws, const float* __restrict__ bs,
    const float* __restrict__ We, const float* __restrict__ be,
    const float* __restrict__ g,  const float* __restrict__ beta,
    float* __restrict__ xn)
{
    const int lane = threadIdx.x & 31;
    const int wave = threadIdx.x >> 5;
    const int row  = blockIdx.x * 8 + wave;       // 3136 blocks * 8 = 25088

    const float* xr = x + (size_t)row * DMODEL;

    float xv[16];
    #pragma unroll
    for (int t = 0; t < 16; ++t) xv[t] = xr[t * 32 + lane];

    float ca[8] = {0.f, 0.f, 0.f, 0.f, 0.f, 0.f, 0.f, 0.f};
    float sa[4] = {0.f, 0.f, 0.f, 0.f};
    #pragma unroll
    for (int t = 0; t < 16; ++t) {
        const int d = t * 32 + lane;
        const float4 w0 = *(const float4*)(Wc + (size_t)d * DCOORD);
        const float4 w1 = *(const float4*)(Wc + (size_t)d * DCOORD + 4);
        ca[0] += xv[t] * w0.x; ca[1] += xv[t] * w0.y;
        ca[2] += xv[t] * w0.z; ca[3] += xv[t] * w0.w;
        ca[4] += xv[t] * w1.x; ca[5] += xv[t] * w1.y;
        ca[6] += xv[t] * w1.z; ca[7] += xv[t] * w1.w;
        const float4 ww = *(const float4*)(Ws + (size_t)d * DSPIN);
        sa[0] += xv[t] * ww.x; sa[1] += xv[t] * ww.y;
        sa[2] += xv[t] * ww.z; sa[3] += xv[t] * ww.w;
    }
    #pragma unroll
    for (int off = 16; off >= 1; off >>= 1) {
        #pragma unroll
        for (int c = 0; c < 8; ++c) ca[c] += __shfl_xor(ca[c], off, 32);
        #pragma unroll
        for (int s = 0; s < 4; ++s) sa[s] += __shfl_xor(sa[s], off, 32);
    }

    float coords[8];
    #pragma unroll
    for (int c = 0; c < 8; ++c) coords[c] = ca[c] + bc[c];
    float K0 = 0.f;
    #pragma unroll
    for (int s = 0; s < 4; ++s) { const float v = sa[s] + bs[s]; K0 += v * v; }

    float xe[16];
    #pragma unroll
    for (int t = 0; t < 16; ++t) {
        const int d = t * 32 + lane;
        float m = be[d];
        #pragma unroll
        for (int c = 0; c < 8; ++c) m += coords[c] * We[c * DMODEL + d];
        xe[t] = m * K0 + xv[t];
    }

    float s = 0.f;
    #pragma unroll
    for (int t = 0; t < 16; ++t) s += xe[t];
    #pragma unroll
    for (int off = 16; off >= 1; off >>= 1) s += __shfl_xor(s, off, 32);
    const float mu = s * (1.0f / (float)DMODEL);

    float v = 0.f;
    #pragma unroll
    for (int t = 0; t < 16; ++t) { const float d0 = xe[t] - mu; v += d0 * d0; }
    #pragma unroll
    for (int off = 16; off >= 1; off >>= 1) v += __shfl_xor(v, off, 32);
    const float rinv = rsqrtf(v * (1.0f / (float)DMODEL) + LN_EPS);

    #pragma unroll
    for (int t = 0; t < 16; ++t) {
        const int d = t * 32 + lane;
        xn[(size_t)row * DMODEL + d] = (xe[t] - mu) * rinv * g[d] + beta[d];
    }
}

// ---------------------------------------------------------------------------
// Kernel 3: out = x_norm @ W_out + b_out (32x32 tile per wave, 2x2 WMMA)
// ---------------------------------------------------------------------------
__global__ __launch_bounds__(128) void out_proj_wmma(
    const float* __restrict__ xn, const float* __restrict__ Wo,
    const float* __restrict__ bo, float* __restrict__ out)
{
    const int lane  = threadIdx.x & 31;
    const int wave  = threadIdx.x >> 5;
    const int tileN = blockIdx.x;                 // 0..15
    const int tileM = blockIdx.y * 4 + wave;      // 0..783
    const int l16   = lane & 15;
    const int khalf = lane >> 4;

    const int mrow0 = tileM * 32 + l16;
    const int ncol0 = tileN * 32 + l16;
    const int ncol1 = ncol0 + 16;
    const float* arow0 = xn + (size_t)mrow0 * DMODEL;
    const float* arow1 = arow0 + (size_t)16 * DMODEL;

    v8f acc00 = {}, acc01 = {}, acc10 = {}, acc11 = {};
    #pragma unroll 2
    for (int k0 = 0; k0 < DMODEL; k0 += 4) {
        const int ka = k0 + khalf * 2;
        const v2f a0 = *(const v2f*)(arow0 + ka);
        const v2f a1 = *(const v2f*)(arow1 + ka);
        v2f b0, b1;
        b0.x = Wo[ka * DMODEL + ncol0];
        b0.y = Wo[(ka + 1) * DMODEL + ncol0];
        b1.x = Wo[ka * DMODEL + ncol1];
        b1.y = Wo[(ka + 1) * DMODEL + ncol1];

        acc00 = __builtin_amdgcn_wmma_f32_16x16x4_f32(false, a0, false, b0, (short)0, acc00, false, false);
        acc01 = __builtin_amdgcn_wmma_f32_16x16x4_f32(false, a0, false, b1, (short)0, acc01, false, false);
        acc10 = __builtin_amdgcn_wmma_f32_16x16x4_f32(false, a1, false, b0, (short)0, acc10, false, false);
        acc11 = __builtin_amdgcn_wmma_f32_16x16x4_f32(false, a1, false, b1, (short)0, acc11, false, false);
    }

    const float bias0 = bo[ncol0];
    const float bias1 = bo[ncol1];
    #pragma unroll
    for (int r = 0; r < 8; ++r) {
        const int m    = r + 8 * khalf;
        const int row0 = tileM * 32 + m;
        const int row1 = row0 + 16;
        out[(size_t)row0 * DMODEL + ncol0] = acc00[r] + bias0;
        out[(size_t)row0 * DMODEL + ncol1] = acc01[r] + bias1;
        out[(size_t)row1 * DMODEL + ncol0] = acc10[r] + bias0;
        out[(size_t)row1 * DMODEL + ncol1] = acc11[r] + bias1;
    }
}

// ---------------------------------------------------------------------------
extern "C" void kernel_launch(void* const* d_in, const int* in_sizes, int n_in,
                              void* d_out, int out_size, void* d_ws, size_t ws_size,
                              hipStream_t stream) {
    const float* images = (const float*)d_in[0];
    const float* Wp     = (const float*)d_in[1];
    const float* bp     = (const float*)d_in[2];
    const float* pos    = (const float*)d_in[3];
    const float* Wc     = (const float*)d_in[4];
    const float* bc     = (const float*)d_in[5];
    const float* Ws     = (const float*)d_in[6];
    const float* bs     = (const float*)d_in[7];
    const float* We     = (const float*)d_in[8];
    const float* be     = (const float*)d_in[9];
    const float* g      = (const float*)d_in[10];
    const float* beta   = (const float*)d_in[11];
    const float* Wo     = (const float*)d_in[12];
    const float* bo     = (const float*)d_in[13];

    float* x   = (float*)d_ws;          // 25088 x 512 fp32 = 51.4 MB scratch
    float* out = (float*)d_out;

    // 25088 rows = 784 32-row tiles = 196 blocks x 4 waves; 512 = 16 32-col tiles
    patch_embed_wmma<<<dim3(16, 196), 128, 0, stream>>>(images, Wp, bp, pos, x);
    // 25088 rows, one wave each, 8 waves/block  (in-place x -> x_norm)
    manifold_ln<<<dim3(3136), 256, 0, stream>>>(x, Wc, bc, Ws, bs, We, be, g, beta, x);
    out_proj_wmma<<<dim3(16, 196), 128, 0, stream>>>(x, Wo, bo, out);
}